// DynamicPoolLoss_89730456748971
// MI455X (gfx1250) — compile-verified
//
#include <hip/hip_runtime.h>
#include <hip/hip_bf16.h>
#include <math.h>

// ---------------------------------------------------------------------------
// Hierarchical contrastive loss (TS2Vec-style) for B=8, T=1024, D=320, fp32.
// Fused GEMM (V_WMMA_F32_16X16X4_F32) + branch-free online log-sum-exp,
// never materializing the 2T x 2T similarity matrices.
// ---------------------------------------------------------------------------

typedef float v2f __attribute__((ext_vector_type(2)));
typedef float v8f __attribute__((ext_vector_type(8)));

#define D_SZ 320
// Finite sentinel instead of -inf: keeps the online-LSE branch-free and
// NaN-free (sentinel - sentinel == 0, and exp(sentinel - realMax) == 0, so
// phantom terms accumulated while only masked elements were seen vanish as
// soon as one real element arrives; fully-masked rows are never read).
#define SENTINEL (-3.402823466e+38f)

static __device__ __forceinline__ v8f wmma_f32_k4(v2f a, v2f b, v8f c) {
  // D = A(16x4, f32) * B(4x16, f32) + C(16x16, f32)
  return __builtin_amdgcn_wmma_f32_16x16x4_f32(false, a, false, b,
                                               (short)0, c, false, false);
}

// branch-free combine of two (max, sumexp) partial LSE states
static __device__ __forceinline__ void lse_combine(float& M, float& S,
                                                   float M2, float S2) {
  float Mn = fmaxf(M, M2);
  S = S * __expf(M - Mn) + S2 * __expf(M2 - Mn);
  M = Mn;
}

// branch-free per-element online LSE + target pick for one 16x16 C tile
static __device__ __forceinline__ void process_tile(
    const v8f& acc, int gcol, int r0, int hv, int t, int twoT,
    float* runM, float* runS, float* tg) {
#pragma unroll
  for (int v = 0; v < 8; ++v) {
    const int grow = r0 + v + 8 * hv;             // C layout: row = v + 8*half
    const float s = acc[v];
    const bool valid = (gcol < twoT) && (grow < twoT) && (gcol != grow);
    const int partner = (grow < t) ? (grow + t) : (grow - t);
    tg[v] += (gcol == partner && grow < twoT) ? s : 0.0f;
    const float sv = valid ? s : SENTINEL;
    const float mn = fmaxf(runM[v], sv);
    runS[v] = runS[v] * __expf(runM[v] - mn) + __expf(sv - mn);
    runM[v] = mn;
  }
}

// ---------------------------------------------------------------------------
// Temporal loss: for batch b, Z = concat(z1[b], z2[b]) (2t x 320),
// sim = Z Z^T.  One wave per 16-row tile; sweep column tiles two at a time
// (two independent WMMA accumulator chains sharing the LDS A reads) with
// online LSE over off-diagonal entries; target = sim(i, (i+t) mod 2t).
// partials[b*gridDim.x + rowTile] = weight * sum_rows(LSE - target).
// ---------------------------------------------------------------------------
__global__ __launch_bounds__(32)
void hcl_temporal_kernel(const float* __restrict__ z1,
                         const float* __restrict__ z2,
                         int t, float* __restrict__ partials, float weight) {
  const int b    = blockIdx.y;
  const int rb   = blockIdx.x;
  const int lane = threadIdx.x;
  const int hv   = lane >> 4;        // half-wave: 0 or 1
  const int m    = lane & 15;
  const int koff = hv * 2;           // K sub-offset for this half (f32 WMMA)
  const int twoT = 2 * t;
  const int r0   = rb * 16;

  __shared__ float Arows[16 * 322];  // padded stride (322 % 64 == 2): no bank conflicts

  // Stage the 16 A rows (clamped for small t) into LDS, coalesced.
  for (int rr = 0; rr < 16; ++rr) {
    int grow = r0 + rr; if (grow >= twoT) grow = twoT - 1;
    const float* src = (grow < t)
        ? (z1 + ((size_t)b * t + grow) * D_SZ)
        : (z2 + ((size_t)b * t + (grow - t)) * D_SZ);
    for (int kk = lane; kk < D_SZ; kk += 32) Arows[rr * 322 + kk] = src[kk];
  }
  __syncthreads();

  float runM[8], runS[8], tg[8];
#pragma unroll
  for (int v = 0; v < 8; ++v) { runM[v] = SENTINEL; runS[v] = 0.0f; tg[v] = 0.0f; }

  const float* a_base = &Arows[m * 322 + koff];
  const int nct   = (twoT + 15) >> 4;   // column tiles
  const int npair = (nct + 1) >> 1;     // processed two at a time

  for (int cp = 0; cp < npair; ++cp) {
    const int c0a = (cp * 2) * 16;
    const int c0b = c0a + 16;
    const int gcolA = c0a + m;
    const int gcolB = c0b + m;
    int ccolA = (gcolA >= twoT) ? (twoT - 1) : gcolA;   // clamp for loads
    int ccolB = (gcolB >= twoT) ? (twoT - 1) : gcolB;
    const float* bA = ((ccolA < t)
        ? (z1 + ((size_t)b * t + ccolA) * D_SZ)
        : (z2 + ((size_t)b * t + (ccolA - t)) * D_SZ)) + koff;
    const float* bB = ((ccolB < t)
        ? (z1 + ((size_t)b * t + ccolB) * D_SZ)
        : (z2 + ((size_t)b * t + (ccolB - t)) * D_SZ)) + koff;

    v8f acc0 = {0.f, 0.f, 0.f, 0.f, 0.f, 0.f, 0.f, 0.f};
    v8f acc1 = {0.f, 0.f, 0.f, 0.f, 0.f, 0.f, 0.f, 0.f};
#pragma unroll 4
    for (int k0 = 0; k0 < D_SZ; k0 += 4) {   // 2 x 80 interleaved fp32 WMMAs
      v2f av  = { a_base[k0], a_base[k0 + 1] };
      v2f bv0 = { bA[k0], bA[k0 + 1] };
      v2f bv1 = { bB[k0], bB[k0 + 1] };
      acc0 = wmma_f32_k4(av, bv0, acc0);
      acc1 = wmma_f32_k4(av, bv1, acc1);
    }

    process_tile(acc0, gcolA, r0, hv, t, twoT, runM, runS, tg);
    process_tile(acc1, gcolB, r0, hv, t, twoT, runM, runS, tg);
  }

  // Reduce each row's LSE state across its 16 lanes (xor masks stay in-half).
  float blockSum = 0.0f;
#pragma unroll
  for (int v = 0; v < 8; ++v) {
    float M = runM[v], S = runS[v], T0 = tg[v];
    for (int o = 1; o < 16; o <<= 1) {
      float M2 = __shfl_xor(M,  o, 32);
      float S2 = __shfl_xor(S,  o, 32);
      float T2 = __shfl_xor(T0, o, 32);
      lse_combine(M, S, M2, S2);
      T0 += T2;
    }
    const int grow = r0 + v + 8 * hv;
    if (m == 0 && grow < twoT) blockSum += (M + logf(S)) - T0;
  }
  blockSum += __shfl_xor(blockSum, 16, 32);        // lanes 0 and 16 combine
  if (lane == 0) partials[b * gridDim.x + rb] = blockSum * weight;
}

// ---------------------------------------------------------------------------
// Instance loss: per time step s, Z_s rows 0..7 = z1[:, s], 8..15 = z2[:, s].
// sim = Z_s Z_s^T (16x16), single tile; A and B operands are identical per
// lane because (m == n == lane%16) with the same K sub-offset.
// partials[s] = weight * sum_rows(LSE_offdiag - sim(i, (i+8) mod 16)).
// ---------------------------------------------------------------------------
__global__ __launch_bounds__(32)
void hcl_instance_kernel(const float* __restrict__ z1,
                         const float* __restrict__ z2,
                         int t, float* __restrict__ partials, float weight) {
  const int s    = blockIdx.x;
  const int lane = threadIdx.x;
  const int hv   = lane >> 4;
  const int m    = lane & 15;
  const int koff = hv * 2;

  const float* row = (m < 8)
      ? (z1 + ((size_t)m * t + s) * D_SZ)
      : (z2 + ((size_t)(m - 8) * t + s) * D_SZ);
  const float* rb = row + koff;

  v8f acc = {0.f, 0.f, 0.f, 0.f, 0.f, 0.f, 0.f, 0.f};
#pragma unroll 8
  for (int k0 = 0; k0 < D_SZ; k0 += 4) {
    v2f av = { rb[k0], rb[k0 + 1] };
    acc = wmma_f32_k4(av, av, acc);               // symmetric: B operand == A
  }

  float blockSum = 0.0f;
#pragma unroll
  for (int v = 0; v < 8; ++v) {
    const int grow    = v + 8 * hv;
    const int partner = (grow + 8) & 15;
    const float sv    = acc[v];
    float M  = (m != grow)    ? sv : SENTINEL;    // drop diagonal
    float S  = (m != grow)    ? 1.0f : 0.0f;
    float T0 = (m == partner) ? sv : 0.0f;
    for (int o = 1; o < 16; o <<= 1) {
      float M2 = __shfl_xor(M,  o, 32);
      float S2 = __shfl_xor(S,  o, 32);
      float T2 = __shfl_xor(T0, o, 32);
      lse_combine(M, S, M2, S2);
      T0 += T2;
    }
    if (m == 0) blockSum += (M + logf(S)) - T0;
  }
  blockSum += __shfl_xor(blockSum, 16, 32);
  if (lane == 0) partials[s] = blockSum * weight;
}

// ---------------------------------------------------------------------------
// Pooling / selection (matches jnp semantics exactly, incl. rank-space gather)
// ---------------------------------------------------------------------------
__global__ void hcl_rank_kernel(const int* __restrict__ time, int t,
                                int* __restrict__ rank) {
  const int b = blockIdx.x;
  const int n = t - 1;
  __shared__ int sh[1024];
  for (int i = threadIdx.x; i < t; i += blockDim.x) sh[i] = time[b * t + i];
  __syncthreads();
  for (int p = threadIdx.x; p < n; p += blockDim.x) {
    const int dp = sh[p + 1] - sh[p];
    int r = 0;
    for (int q = 0; q < n; ++q) {
      const int dq = sh[q + 1] - sh[q];
      r += (dq < dp) || (dq == dp && q < p);      // stable argsort rank
    }
    rank[b * n + p] = r;
  }
}

// Reference does pooled[ sorted rank-positions j where indices[j] < t/2 ]:
// output slot k sources pooled[rank[p]] with slot = #{p'<t/2 : rank[p']<rank[p]}.
__global__ void hcl_select_kernel(const int* __restrict__ rank, int t,
                                  int* __restrict__ selsrc) {
  const int b = blockIdx.x;
  const int h = t >> 1;
  __shared__ int rsh[512];
  for (int p = threadIdx.x; p < h; p += blockDim.x) rsh[p] = rank[b * (t - 1) + p];
  __syncthreads();
  for (int p = threadIdx.x; p < h; p += blockDim.x) {
    const int rp = rsh[p];
    int slot = 0;
    for (int q = 0; q < h; ++q) slot += (rsh[q] < rp);
    selsrc[b * h + slot] = rp;                    // ranks are distinct: bijection
  }
}

__global__ void hcl_pool_kernel(const float* __restrict__ z1,
                                const float* __restrict__ z2,
                                const int* __restrict__ time,
                                const int* __restrict__ selsrc, int t,
                                float* __restrict__ o1, float* __restrict__ o2,
                                int* __restrict__ otime) {
  const int h = t >> 1;
  const int k = blockIdx.x;
  const int b = blockIdx.y;
  const int j = selsrc[b * h + k];                // index into pooled (0..t-2)
  const float* a1 = z1 + ((size_t)b * t + j) * D_SZ;
  const float* a2 = z2 + ((size_t)b * t + j) * D_SZ;
  float* d1 = o1 + ((size_t)b * h + k) * D_SZ;
  float* d2 = o2 + ((size_t)b * h + k) * D_SZ;
  for (int c = threadIdx.x; c < D_SZ; c += blockDim.x) {
    d1[c] = fmaxf(a1[c], a1[c + D_SZ]);
    d2[c] = fmaxf(a2[c], a2[c + D_SZ]);
  }
  if (threadIdx.x == 0)
    otime[b * h + k] = (time[b * t + j] + time[b * t + j + 1]) / 2;
}

// Fixed-order deterministic final reduction.
__global__ void hcl_reduce_kernel(const float* __restrict__ partials, int n,
                                  float* __restrict__ out) {
  if (threadIdx.x == 0 && blockIdx.x == 0) {
    float s = 0.0f;
    for (int i = 0; i < n; ++i) s += partials[i];
    out[0] = s;
  }
}

// ---------------------------------------------------------------------------
extern "C" void kernel_launch(void* const* d_in, const int* in_sizes, int n_in,
                              void* d_out, int out_size, void* d_ws, size_t ws_size,
                              hipStream_t stream) {
  const float* z1 = (const float*)d_in[0];   // (8, 1024, 320) f32
  const float* z2 = (const float*)d_in[1];   // (8, 1024, 320) f32
  const int* etim = (const int*)d_in[2];     // (8, 1024) i32

  // workspace layout
  char* ws = (char*)d_ws;
  size_t off = 0;
  auto take = [&](size_t bytes) -> void* {
    void* p = (void*)(ws + off);
    off += (bytes + 255) & ~((size_t)255);
    return p;
  };
  float* partials = (float*)take(8192 * sizeof(float));
  int* d_rank     = (int*)take(8 * 1023 * sizeof(int));
  int* d_sel      = (int*)take(8 * 512 * sizeof(int));
  int* tA         = (int*)take(8 * 512 * sizeof(int));
  int* tB         = (int*)take(8 * 512 * sizeof(int));
  const size_t zb = (size_t)8 * 512 * D_SZ * sizeof(float);
  float* z1A = (float*)take(zb); float* z2A = (float*)take(zb);
  float* z1B = (float*)take(zb); float* z2B = (float*)take(zb);
  (void)ws_size; (void)in_sizes; (void)n_in; (void)out_size;

  const float* cz1 = z1; const float* cz2 = z2; const int* ct = etim;
  int t = 1024;
  int slotOff = 0;

  for (int lvl = 0; lvl < 10; ++lvl) {           // t = 1024 >> lvl, down to 2
    // alpha = 0.5, (1-alpha) = 0.5, final /11; both losses are means over
    // B*2t = 16t rows, so both per-row weights coincide:
    const float w = 0.5f / (16.0f * (float)t * 11.0f);

    hcl_instance_kernel<<<t, 32, 0, stream>>>(cz1, cz2, t, partials + slotOff, w);
    slotOff += t;

    const int nrt = (2 * t + 15) >> 4;
    hcl_temporal_kernel<<<dim3(nrt, 8), 32, 0, stream>>>(cz1, cz2, t,
                                                         partials + slotOff, w);
    slotOff += nrt * 8;

    // pool + filter to t/2
    hcl_rank_kernel<<<8, 256, 0, stream>>>(ct, t, d_rank);
    hcl_select_kernel<<<8, 256, 0, stream>>>(d_rank, t, d_sel);
    const int h = t >> 1;
    float* n1 = (lvl & 1) ? z1B : z1A;
    float* n2 = (lvl & 1) ? z2B : z2A;
    int*   nt = (lvl & 1) ? tB  : tA;
    hcl_pool_kernel<<<dim3(h, 8), 128, 0, stream>>>(cz1, cz2, ct, d_sel, t,
                                                    n1, n2, nt);
    cz1 = n1; cz2 = n2; ct = nt; t = h;
  }

  // final level: t == 1, instance loss only
  hcl_instance_kernel<<<1, 32, 0, stream>>>(cz1, cz2, 1, partials + slotOff,
                                            0.5f / (16.0f * 11.0f));
  slotOff += 1;

  hcl_reduce_kernel<<<1, 32, 0, stream>>>(partials, slotOff, (float*)d_out);
}